// myloss_4982162063510
// MI455X (gfx1250) — compile-verified
//
#include <hip/hip_runtime.h>
#include <stdint.h>

// Weighted cross-entropy: loss = sum_i (logsumexp(predict_i) - predict_i[y_i]) * w_i / B
// B = 8192 rows, V = 32000 cols, f32. Pure HBM-streaming reduction:
// 1.05 GB @ 23.3 TB/s => ~45us floor. Strategy: stage each full row (125 KB)
// into CDNA5's 320 KB LDS with async global->LDS B128 copies (ASYNCcnt path),
// then two LDS passes (max, sum-of-exp) with exactly one v_exp_f32 per element.

#define THREADS 256
#define MAX_V   32000
#define NWAVES  (THREADS / 32)

typedef int v4i __attribute__((ext_vector_type(4)));
typedef __attribute__((address_space(1))) v4i* gas_v4i_ptr;
typedef __attribute__((address_space(3))) v4i* las_v4i_ptr;

// ---- gfx1250 async global->LDS copy (16B per lane), with safe fallback ----
__device__ __forceinline__ void async_copy_b128(const float* gsrc, float* ldst) {
#if defined(__has_builtin) && __has_builtin(__builtin_amdgcn_global_load_async_to_lds_b128)
  __builtin_amdgcn_global_load_async_to_lds_b128(
      (gas_v4i_ptr)(v4i*)const_cast<float*>(gsrc),
      (las_v4i_ptr)(v4i*)ldst,
      /*imm offset*/ 0, /*cpol*/ 0);
#else
  *(float4*)ldst = *(const float4*)gsrc;   // fallback: copy through VGPRs
#endif
}

__device__ __forceinline__ void wait_async_all() {
#if defined(__has_builtin) && __has_builtin(__builtin_amdgcn_s_wait_asynccnt)
  __builtin_amdgcn_s_wait_asynccnt(0);
#elif defined(__gfx1250__)
  asm volatile("s_wait_asynccnt 0x0" ::: "memory");
#endif
}

__global__ __launch_bounds__(THREADS)
void ce_row_kernel(const float* __restrict__ predict,
                   const int*   __restrict__ y,
                   const float* __restrict__ weight,
                   float*       __restrict__ row_loss,
                   int V)
{
  __shared__ float row[MAX_V];
  __shared__ float red[NWAVES];

  const int row_id = blockIdx.x;
  const int t      = threadIdx.x;
  const float* g   = predict + (size_t)row_id * (size_t)V;

  // ---- Stage the whole row into LDS via async DMA-style copies ----
  const int nchunk = V >> 2;                       // float4 chunks
  for (int c = t; c < nchunk; c += THREADS)
    async_copy_b128(g + 4 * c, &row[4 * c]);
  const int tail = V & 3;                          // scalar tail (V % 4)
  if (t < tail) row[(nchunk << 2) + t] = g[(nchunk << 2) + t];

  wait_async_all();       // this wave's async copies landed in LDS
  __syncthreads();        // all waves' copies landed

  // ---- Pass 1: row max (stride-THREADS LDS reads: bank-conflict free) ----
  float m = -__builtin_inff();
  for (int e = t; e < V; e += THREADS) m = fmaxf(m, row[e]);
  #pragma unroll
  for (int off = 16; off > 0; off >>= 1)
    m = fmaxf(m, __shfl_xor(m, off, 32));          // wave32 butterfly
  if ((t & 31) == 0) red[t >> 5] = m;
  __syncthreads();
  float M = red[0];
  #pragma unroll
  for (int w = 1; w < NWAVES; ++w) M = fmaxf(M, red[w]);
  __syncthreads();                                 // red about to be reused

  // ---- Pass 2: sum of exp(x - M): one v_exp_f32 per element ----
  float s = 0.0f;
  for (int e = t; e < V; e += THREADS) s += __expf(row[e] - M);
  #pragma unroll
  for (int off = 16; off > 0; off >>= 1)
    s += __shfl_xor(s, off, 32);
  if ((t & 31) == 0) red[t >> 5] = s;
  __syncthreads();

  if (t == 0) {
    float S = 0.0f;
    #pragma unroll
    for (int w = 0; w < NWAVES; ++w) S += red[w];
    const float lse = M + __logf(S);
    const float tgt = row[y[row_id]];              // target logit straight from LDS
    row_loss[row_id] = (lse - tgt) * weight[row_id];
  }
}

// ---- Deterministic final reduction: sum(row_loss) / B -> out[0] ----
__global__ __launch_bounds__(256)
void ce_finalize_kernel(const float* __restrict__ row_loss,
                        float* __restrict__ out, int B)
{
  __shared__ float red[8];
  const int t = threadIdx.x;
  float s = 0.0f;
  for (int i = t; i < B; i += 256) s += row_loss[i];
  #pragma unroll
  for (int off = 16; off > 0; off >>= 1)
    s += __shfl_xor(s, off, 32);
  if ((t & 31) == 0) red[t >> 5] = s;
  __syncthreads();
  if (t == 0) {
    float S = 0.0f;
    #pragma unroll
    for (int w = 0; w < 8; ++w) S += red[w];
    out[0] = S / (float)B;
  }
}

extern "C" void kernel_launch(void* const* d_in, const int* in_sizes, int n_in,
                              void* d_out, int out_size, void* d_ws, size_t ws_size,
                              hipStream_t stream) {
  const float* predict = (const float*)d_in[0];
  const int*   y       = (const int*)d_in[1];   // harness delivers integer inputs as int32
  const float* weight  = (const float*)d_in[2];

  const int B = in_sizes[1];                    // 8192
  const int V = in_sizes[0] / B;                // 32000

  float* row_loss = (float*)d_ws;               // B floats of scratch

  ce_row_kernel<<<B, THREADS, 0, stream>>>(predict, y, weight, row_loss, V);
  ce_finalize_kernel<<<1, 256, 0, stream>>>(row_loss, (float*)d_out, B);
}